// HashNgramEmbeddings_12549894439058
// MI455X (gfx1250) — compile-verified
//
#include <hip/hip_runtime.h>
#include <cstdint>

// ---------------------------------------------------------------------------
// HashNgramEmbeddings on MI455X (gfx1250)
//
// Memory-bound gather: ~96MB random 1KB-row reads + 32MB writes ~= 5.5us at
// 23.3 TB/s. Strategy:
//   * emb3/emb4 rows: CDNA5 Tensor Data Mover in GATHER MODE (8 rows per
//     descriptor via 32-bit indices in D# groups 2/3) -> LDS, TENSORcnt.
//   * emb2 rows: plain global_load_b128 with wave-uniform SGPR bases, issued
//     before s_wait_tensorcnt so the VMEM and TDM paths stream in parallel.
//   * output: nontemporal stores (streamed once; keep L2 for table rows).
// ---------------------------------------------------------------------------

typedef unsigned int v4u __attribute__((ext_vector_type(4)));
typedef int          v4i __attribute__((ext_vector_type(4)));
typedef int          v8i __attribute__((ext_vector_type(8)));
typedef float        v4f __attribute__((ext_vector_type(4)));

#define NGRAM_VOCAB 500000
#define EMB_D 256
#define SEQ_T 4096
#define BATCH 8
#define NT 8                 // t-positions per wave = max 32-bit gather indices per D#
#define WAVES_PER_BLOCK 2
#define ROW_BYTES (EMB_D * 4)

// Issue one TDM gather of NT rows (32-bit indices) from `table` into LDS at
// byte offset lds_off. Descriptor layout per CDNA5 ISA ch. 8 (async/tensor).
__device__ __forceinline__ void tdm_gather8(unsigned lds_off,
                                            const float* table,
                                            v4i idx_lo, v4i idx_hi) {
  const unsigned long long ga = (unsigned long long)(uintptr_t)table;
  // Group 0: count=1 | gather_index_size=32b (bit30) | gather_mode (bit31);
  //          lds_addr; global_addr[56:0]; type=2 ("image") in bits 127:126.
  v4u g0 = { 0xC0000001u,
             lds_off,
             (unsigned)(ga & 0xFFFFFFFFu),
             (unsigned)((ga >> 32) & 0x01FFFFFFu) | (2u << 30) };
  // Group 1: workgroup_mask=0 (not in cluster), data_size=2 (4B),
  //          tensor_dim0=256, tensor_dim1=500000 (row OOB bound),
  //          tile_dim0=256 (row width), tile_dim1=8 (#valid indices),
  //          tensor_dim0_stride=256 elements. Padding/iterate disabled.
  v8i g1 = { (int)0x00020000,                       // data_size=4B
             (int)(EMB_D << 16),                    // tensor_dim0[15:0]
             (int)((NGRAM_VOCAB & 0xFFFF) << 16),   // tensor_dim1[15:0]
             (int)((NGRAM_VOCAB >> 16) | (EMB_D << 16)), // dim1 hi | tile_dim0
             (int)NT,                               // tile_dim1 = #indices
             (int)EMB_D,                            // tensor_dim0_stride
             0, 0 };
#if defined(__clang_major__) && __clang_major__ >= 23
  v8i zero8 = {0, 0, 0, 0, 0, 0, 0, 0};
  __builtin_amdgcn_tensor_load_to_lds(g0, g1, idx_lo, idx_hi, zero8, 0);
#else
  __builtin_amdgcn_tensor_load_to_lds(g0, g1, idx_lo, idx_hi, 0);
#endif
}

__global__ void __launch_bounds__(WAVES_PER_BLOCK * 32)
hash_ngram_gather_kernel(const int* __restrict__ byte_ids,
                         const float* __restrict__ emb2,
                         const float* __restrict__ emb3,
                         const float* __restrict__ emb4,
                         float* __restrict__ out) {
  // Per wave: 2 TDM-gathered tables x 8 rows x 1KB = 16KB. 2 waves -> 32KB.
  __shared__ float smem[WAVES_PER_BLOCK * 2 * NT * EMB_D];

  const int tid  = threadIdx.x;
  const int wave = tid >> 5;
  const int lane = tid & 31;

  const int chunk = blockIdx.x * WAVES_PER_BLOCK + wave;  // 4096 chunks total
  const int b     = chunk >> 9;                           // 512 chunks per row
  const int tbase = (chunk & 511) * NT;

  // --- per-lane polynomial hash for t = tbase + (lane & 7) ---------------
  // h_n(t) = sum_i byte[t-n+1+i] * 31^(n-1-i) mod 500000
  const int j = lane & 7;
  const int t = tbase + j;
  const int rowoff = b * SEQ_T;
  const int p0 = byte_ids[rowoff + t];
  const int p1 = byte_ids[rowoff + (t >= 1 ? t - 1 : 0)];
  const int p2 = byte_ids[rowoff + (t >= 2 ? t - 2 : 0)];
  const int p3 = byte_ids[rowoff + (t >= 3 ? t - 3 : 0)];
  const int h2 = (p1 * 31 + p0) % NGRAM_VOCAB;
  const int h3 = (p2 * 961 + p1 * 31 + p0) % NGRAM_VOCAB;
  const int h4 = (p3 * 29791 + p2 * 961 + p1 * 31 + p0) % NGRAM_VOCAB;

  // --- broadcast 8 indices per table into wave-uniform scalars -----------
  v4i i2lo = { __builtin_amdgcn_readlane(h2, 0), __builtin_amdgcn_readlane(h2, 1),
               __builtin_amdgcn_readlane(h2, 2), __builtin_amdgcn_readlane(h2, 3) };
  v4i i2hi = { __builtin_amdgcn_readlane(h2, 4), __builtin_amdgcn_readlane(h2, 5),
               __builtin_amdgcn_readlane(h2, 6), __builtin_amdgcn_readlane(h2, 7) };
  v4i i3lo = { __builtin_amdgcn_readlane(h3, 0), __builtin_amdgcn_readlane(h3, 1),
               __builtin_amdgcn_readlane(h3, 2), __builtin_amdgcn_readlane(h3, 3) };
  v4i i3hi = { __builtin_amdgcn_readlane(h3, 4), __builtin_amdgcn_readlane(h3, 5),
               __builtin_amdgcn_readlane(h3, 6), __builtin_amdgcn_readlane(h3, 7) };
  v4i i4lo = { __builtin_amdgcn_readlane(h4, 0), __builtin_amdgcn_readlane(h4, 1),
               __builtin_amdgcn_readlane(h4, 2), __builtin_amdgcn_readlane(h4, 3) };
  v4i i4hi = { __builtin_amdgcn_readlane(h4, 4), __builtin_amdgcn_readlane(h4, 5),
               __builtin_amdgcn_readlane(h4, 6), __builtin_amdgcn_readlane(h4, 7) };

  // --- issue 2 TDM gathers (8 rows x 1KB each) into this wave's LDS ------
  const unsigned ldsbase = (unsigned)(wave * 2 * NT * ROW_BYTES);
  tdm_gather8(ldsbase + 0u * NT * ROW_BYTES, emb3, i3lo, i3hi);
  tdm_gather8(ldsbase + 1u * NT * ROW_BYTES, emb4, i4lo, i4hi);

  // --- overlap: direct VMEM gather of emb2 rows (SGPR base + lane off) ---
  const unsigned h2s[NT] = { (unsigned)i2lo.x, (unsigned)i2lo.y,
                             (unsigned)i2lo.z, (unsigned)i2lo.w,
                             (unsigned)i2hi.x, (unsigned)i2hi.y,
                             (unsigned)i2hi.z, (unsigned)i2hi.w };
  v4f e2a[NT], e2b[NT];
#pragma unroll
  for (int jj = 0; jj < NT; ++jj) {
    const v4f* p = (const v4f*)(emb2 + (size_t)h2s[jj] * EMB_D + lane * 8);
    e2a[jj] = p[0];
    e2b[jj] = p[1];
  }

  __builtin_amdgcn_s_wait_tensorcnt(0);

  // The TDM wrote smem behind the compiler's back; let the address escape
  // into an asm with a memory clobber so the LDS loads below are not folded,
  // while keeping the actual reads on the addrspace(3) path (ds_load_b128).
  {
    float* p = &smem[0];
    asm volatile("" : : "v"(p) : "memory");
  }

  const int w3base = wave * 2 * NT * EMB_D + 0 * NT * EMB_D;
  const int w4base = wave * 2 * NT * EMB_D + 1 * NT * EMB_D;

  // --- accumulate 3 rows per t (masking the left-pad) and store ----------
#pragma unroll
  for (int jj = 0; jj < NT; ++jj) {
    const int tt = tbase + jj;
    const float m2 = (tt >= 1) ? 1.0f : 0.0f;
    const float m3 = (tt >= 2) ? 1.0f : 0.0f;
    const float m4 = (tt >= 3) ? 1.0f : 0.0f;
    const int e = jj * EMB_D + lane * 8;           // 8 floats per lane
    const v4f r3a = *(const v4f*)&smem[w3base + e];
    const v4f r3b = *(const v4f*)&smem[w3base + e + 4];
    const v4f r4a = *(const v4f*)&smem[w4base + e];
    const v4f r4b = *(const v4f*)&smem[w4base + e + 4];
    v4f a0 = 0.25f * (m2 * e2a[jj] + m3 * r3a + m4 * r4a);
    v4f a1 = 0.25f * (m2 * e2b[jj] + m3 * r3b + m4 * r4b);
    v4f* o = (v4f*)&out[((size_t)b * SEQ_T + tt) * EMB_D + lane * 8];
    __builtin_nontemporal_store(a0, o);
    __builtin_nontemporal_store(a1, o + 1);
  }
}

extern "C" void kernel_launch(void* const* d_in, const int* in_sizes, int n_in,
                              void* d_out, int out_size, void* d_ws, size_t ws_size,
                              hipStream_t stream) {
  (void)in_sizes; (void)n_in; (void)out_size; (void)d_ws; (void)ws_size;
  const int*   byte_ids = (const int*)d_in[0];
  const float* emb2     = (const float*)d_in[1];
  const float* emb3     = (const float*)d_in[2];
  const float* emb4     = (const float*)d_in[3];
  float*       out      = (float*)d_out;

  const int chunks = BATCH * (SEQ_T / NT);          // 4096
  dim3 grid(chunks / WAVES_PER_BLOCK);              // 2048 blocks
  dim3 block(WAVES_PER_BLOCK * 32);                 // 64 threads = 2 waves
  hipLaunchKernelGGL(hash_ngram_gather_kernel, grid, block, 0, stream,
                     byte_ids, emb2, emb3, emb4, out);
}